// FALayer_49134425866991
// MI455X (gfx1250) — compile-verified
//
#include <hip/hip_runtime.h>
#include <math.h>

typedef __attribute__((ext_vector_type(2))) float v2f;
typedef __attribute__((ext_vector_type(4))) float v4f;
typedef __attribute__((ext_vector_type(8))) float v8f;

#define NDIM 64

// ---------------------------------------------------------------------------
// Kernel A: gd = h @ W_dst, gs = h @ W_src via V_WMMA_F32_16X16X4_F32.
// One wave32 handles 16 nodes. B-matrix column 0 = W_dst, column 1 = W_src,
// remaining 14 columns zero. K=64 consumed in 16 steps of K=4.
// B fragments are loop-invariant -> preloaded branch-free into 16 v2f regs so
// the inner loop is a clean global_load_b64(A) + v_wmma stream (no exec churn).
// ---------------------------------------------------------------------------
__global__ void fa_gate_wmma(const float* __restrict__ h,
                             const float* __restrict__ gate_W,
                             float* __restrict__ gd,
                             float* __restrict__ gs,
                             int N)
{
    int wave = (blockIdx.x * blockDim.x + threadIdx.x) >> 5;
    int lane = threadIdx.x & 31;
    int base = wave * 16;
    if (base >= N) return;                 // wave-uniform: EXEC stays all-1s

    int row   = lane & 15;                 // A-row / B-column index
    int khalf = (lane >> 4) << 1;          // 0 for lanes 0-15, 2 for 16-31
    int arow  = base + row; if (arow > N - 1) arow = N - 1;   // tail clamp

    // Preload B: col 0 = W_dst, col 1 = W_src, cols 2..15 = 0 (mask-multiply,
    // no divergence; all lanes read the same 128-float table -> broadcast).
    float msk = (row < 2) ? 1.0f : 0.0f;
    const float* wbase = gate_W + ((row < 2) ? row : 0) * NDIM + khalf;
    v2f bfrag[16];
    #pragma unroll
    for (int j = 0; j < 16; ++j) {
        bfrag[j].x = wbase[j * 4]     * msk;
        bfrag[j].y = wbase[j * 4 + 1] * msk;
    }

    const float* arow_p = h + (size_t)arow * NDIM + khalf;
    v8f acc = {0.f, 0.f, 0.f, 0.f, 0.f, 0.f, 0.f, 0.f};
    #pragma unroll
    for (int j = 0; j < 16; ++j) {
        v2f a = *(const v2f*)(arow_p + j * 4);
        acc = __builtin_amdgcn_wmma_f32_16x16x4_f32(
                  /*neg_a=*/false, a, /*neg_b=*/false, bfrag[j],
                  /*c_mod=*/(short)0, acc, /*reuse_a=*/false, /*reuse_b=*/false);
    }

    // D[m][0] = gd(node m) lives in lane N=0/16; D[m][1] = gs in lane N=1/17.
    if (row < 2) {
        float* outp = (row == 0) ? gd : gs;
        int moff = (lane >> 4) << 3;        // lanes 16-31 hold rows 8..15
        #pragma unroll
        for (int i = 0; i < 8; ++i) {
            int m = base + moff + i;
            if (m < N) outp[m] = acc[i];
        }
    }
}

// ---------------------------------------------------------------------------
// Kernel B: per-edge gate coefficient e = tanh(gd[dst]+gs[src]+b)*d[dst]*d[src]
// One thread per edge; edge-list reads fully coalesced, the four scalar
// gathers hit L2 (gd/gs/d total ~1.2 MB).
// ---------------------------------------------------------------------------
__global__ void fa_edge_gate(const float* __restrict__ gd,
                             const float* __restrict__ gs,
                             const float* __restrict__ dvec,
                             const float* __restrict__ gate_b,
                             const int* __restrict__ esrc,
                             const int* __restrict__ edst,
                             float* __restrict__ e, int E)
{
    int i = blockIdx.x * blockDim.x + threadIdx.x;
    if (i >= E) return;
    int s = esrc[i];
    int t = edst[i];
    float g = tanhf(gd[t] + gs[s] + gate_b[0]);
    e[i] = g * dvec[t] * dvec[s];
}

// ---------------------------------------------------------------------------
// Kernel C: zero the output accumulator (harness poisons d_out with 0xAA).
// ---------------------------------------------------------------------------
__global__ void fa_zero(float* __restrict__ z, int n4)
{
    int i = blockIdx.x * blockDim.x + threadIdx.x;
    if (i < n4) ((v4f*)z)[i] = (v4f){0.f, 0.f, 0.f, 0.f};
}

// ---------------------------------------------------------------------------
// Kernel D: z[dst] += e * h[src]. One wave32 per edge, lane owns features
// 2*lane, 2*lane+1: one global_load_b64 of h[src] + two no-return
// global_atomic_add_f32 into z[dst]. h (25.6 MB) and z (25.6 MB) both live in
// the 192 MB L2, so this is L2/atomic-throughput bound; src/dst/e scalar reads
// are wave-uniform broadcasts.
// ---------------------------------------------------------------------------
__global__ void fa_scatter(const float* __restrict__ h,
                           const int* __restrict__ esrc,
                           const int* __restrict__ edst,
                           const float* __restrict__ e,
                           float* __restrict__ z, int E)
{
    int lane   = threadIdx.x & 31;
    int wave   = (blockIdx.x * blockDim.x + threadIdx.x) >> 5;
    int nwaves = (gridDim.x * blockDim.x) >> 5;
    int f = lane << 1;

    for (int i = wave; i < E; i += nwaves) {
        int   s = esrc[i];
        int   t = edst[i];
        float c = e[i];
        v2f hv = *(const v2f*)(h + (size_t)s * NDIM + f);
        float* zp = z + (size_t)t * NDIM + f;
        atomicAdd(zp,     hv.x * c);   // result unused -> no-return atomic
        atomicAdd(zp + 1, hv.y * c);
    }
}

// ---------------------------------------------------------------------------
// Host-side launcher.
// Inputs: 0:h (N*64 f32) 1:d (N f32) 2:gate_W (128 f32) 3:gate_b (1 f32)
//         4:edge_src (E i32) 5:edge_dst (E i32)
// Output: z (N*64 f32).  Workspace: gd[N] | gs[N] | e[E]  (~13.6 MB).
// ---------------------------------------------------------------------------
extern "C" void kernel_launch(void* const* d_in, const int* in_sizes, int n_in,
                              void* d_out, int out_size, void* d_ws, size_t ws_size,
                              hipStream_t stream)
{
    const float* h      = (const float*)d_in[0];
    const float* dvec   = (const float*)d_in[1];
    const float* gate_W = (const float*)d_in[2];
    const float* gate_b = (const float*)d_in[3];
    const int*   esrc   = (const int*)d_in[4];
    const int*   edst   = (const int*)d_in[5];

    int N = in_sizes[1];
    int E = in_sizes[4];
    float* z = (float*)d_out;

    float* gd = (float*)d_ws;
    float* gs = gd + N;
    float* e  = gs + N;

    // A: WMMA gate mat-vecs (16 nodes / wave)
    {
        int waves   = (N + 15) / 16;
        int threads = waves * 32;
        dim3 blk(256), grd((threads + 255) / 256);
        fa_gate_wmma<<<grd, blk, 0, stream>>>(h, gate_W, gd, gs, N);
    }
    // B: per-edge coefficients
    {
        dim3 blk(256), grd((E + 255) / 256);
        fa_edge_gate<<<grd, blk, 0, stream>>>(gd, gs, dvec, gate_b, esrc, edst, e, E);
    }
    // C: clear output
    {
        int n4 = out_size / 4;
        dim3 blk(256), grd((n4 + 255) / 256);
        fa_zero<<<grd, blk, 0, stream>>>(z, n4);
    }
    // D: wave-per-edge scatter-add (grid-stride over edges)
    {
        dim3 blk(256), grd(25000);   // 200K waves, ~16 edges each
        fa_scatter<<<grd, blk, 0, stream>>>(h, esrc, edst, e, z, E);
    }
}